// LinearInternal_8976481648907
// MI455X (gfx1250) — compile-verified
//
#include <hip/hip_runtime.h>
#include <stdint.h>

typedef __attribute__((ext_vector_type(2))) float v2f;
typedef __attribute__((ext_vector_type(8))) float v8f;
typedef __attribute__((ext_vector_type(4))) unsigned int v4u;
typedef __attribute__((ext_vector_type(4))) int v4i;
typedef __attribute__((ext_vector_type(8))) int v8i;

#define C_CH   128
#define N_LEN  4096
#define B_SZ   64
#define W_OUT  64
#define WIN_K  4033
#define KSPLIT 4
#define KPER   (N_LEN / KSPLIT)   // 1024
#define KCHUNK 64
#define NCHUNKS (KPER / KCHUNK)   // 16

#if __has_builtin(__builtin_amdgcn_tensor_load_to_lds) && \
    __has_builtin(__builtin_amdgcn_s_wait_tensorcnt)
#define USE_TDM 1
#else
#define USE_TDM 0
#endif

// One block per (channel, real/imag, k-split). 64(b) x 64(t) fp32 partial GEMM
// out[b,t] += sum_n x[b,n,c,p] * w[c, n-t] via V_WMMA_F32_16X16X4_F32.
// X tiles staged to LDS by the Tensor Data Mover (double buffered); the banded
// filter window is staged by threads (128 dwords/chunk).
__global__ __launch_bounds__(256)
void corr_gemm_kernel(const float* __restrict__ x,
                      const float* __restrict__ w_in_real,
                      const float* __restrict__ w_in_imag,
                      float* __restrict__ part)
{
    const int c  = blockIdx.x;   // channel
    const int p  = blockIdx.y;   // 0 = real, 1 = imag
    const int ks = blockIdx.z;   // k-split
    const float* __restrict__ w_in = (p == 0) ? w_in_real : w_in_imag;

#if USE_TDM
    __shared__ float Xs[2][B_SZ][68];          // 64 b x 64 n (+4 dword TDM pad)
    __shared__ float Ws[2][KCHUNK + 64 + 4];   // filter band w[c, n0-63 .. n0+64]
#else
    __shared__ float Xs[1][B_SZ][68];
    __shared__ float Ws[1][KCHUNK + 64 + 4];
#endif

    const int tid  = threadIdx.x;
    const int wave = tid >> 5;
    const int lane = tid & 31;
    const int lm   = lane & 15;   // M (A) / N (B,C) index within tile
    const int hi   = lane >> 4;   // half-wave selector (K pairing / M+8)
    const int m_base = (wave & 3) << 4;   // b-tile: 0,16,32,48
    const int t_base = (wave >> 2) << 5;  // t columns: 0..31 or 32..63

    v8f acc0 = {}, acc1 = {};
    const int n_start = ks * KPER;

#if USE_TDM
    // ---- Tensor DMA descriptor (ISA 8.3-8.6). 3D tile:
    //   X: 1 elem (4B, the fixed c/p slot)      -> contiguous unit
    //   Y: 64 n, stride 256 elems (= C*2)       -> tensor_dim0_stride
    //   Z: 64 b, stride 0x100000 elems (= N*C*2)-> tensor_dim1_stride
    // LDS pad: 4 dwords after every 64 dwords -> row stride 68 (bank skew).
    const v8i g1 = { (int)((2u << 16)    // data_size = 4 bytes
                   |       (1u << 20)    // pad_enable
                   |       (5u << 22)    // pad_interval: 64 dwords
                   |       (3u << 25)),  // pad_amount: 4 dwords
                     (int)(1u << 16),            // tensor_dim0 = 1 (bits 79:48)
                     (int)(64u << 16),           // tensor_dim1 = 64 (bits 111:80)
                     (int)(1u << 16),            // tile_dim0 = 1 (bits 127:112)
                     (int)(64u | (64u << 16)),   // tile_dim1 = 64, tile_dim2 = 64
                     256,                        // tensor_dim0_stride (48b, low)
                     0,                          // dim0_stride hi | dim1_stride low16
                     0x10 };                     // tensor_dim1_stride = 0x100000 >> 16
    const v4i g2 = { 64, 0, 0, 0 };              // tensor_dim2 = 64
    const v4i g3 = { 0, 0, 0, 0 };
    const v8i g4 = { 0, 0, 0, 0, 0, 0, 0, 0 };   // 6-arg form: extra group (unused)
    const uint64_t gbase = (uint64_t)(uintptr_t)x +
        ((uint64_t)n_start * (C_CH * 2) + (uint64_t)c * 2 + (uint64_t)p) * 4u;
    const unsigned lds_base[2] = { (unsigned)(uintptr_t)&Xs[0][0][0],
                                   (unsigned)(uintptr_t)&Xs[1][0][0] };

    auto issue_tdm = [&](int chunk, int buf) {
        uint64_t ga = gbase + (uint64_t)chunk * (KCHUNK * C_CH * 2 * 4);
        v4u g0;
        g0.x = 1u;                                   // count = 1 valid descriptor
        g0.y = lds_base[buf];                        // lds_addr
        g0.z = (unsigned)(ga & 0xffffffffu);         // global_addr[31:0]
        g0.w = (unsigned)((ga >> 32) & 0x01ffffffu)  // global_addr[56:32]
             | (2u << 30);                           // type = 2 ("image")
        __builtin_amdgcn_tensor_load_to_lds(g0, g1, g2, g3, g4, 0);
    };
    auto stage_ws = [&](int chunk, int buf) {
        if (tid < 128) {
            int widx = n_start + chunk * KCHUNK - 63 + tid;
            Ws[buf][tid] = (widx >= 0 && widx < WIN_K) ? w_in[c * WIN_K + widx] : 0.0f;
        }
    };

    if (wave == 0) issue_tdm(0, 0);
    stage_ws(0, 0);
    for (int ch = 0; ch < NCHUNKS; ++ch) {
        const int cur = ch & 1;
        if (ch + 1 < NCHUNKS) {
            if (wave == 0) issue_tdm(ch + 1, cur ^ 1);   // prefetch next chunk
            stage_ws(ch + 1, cur ^ 1);
            if (wave == 0) __builtin_amdgcn_s_wait_tensorcnt(1);  // chunk ch landed
        } else {
            if (wave == 0) __builtin_amdgcn_s_wait_tensorcnt(0);
        }
        __syncthreads();
        #pragma unroll
        for (int kk = 0; kk < KCHUNK; kk += 4) {
            const int krow = kk + 2 * hi;
            v2f a = *(const v2f*)&Xs[cur][m_base + lm][krow];
            int j0 = krow - (t_base + lm) + 63;
            v2f b0; b0.x = Ws[cur][j0];      b0.y = Ws[cur][j0 + 1];
            v2f b1; b1.x = Ws[cur][j0 - 16]; b1.y = Ws[cur][j0 - 15];
            acc0 = __builtin_amdgcn_wmma_f32_16x16x4_f32(
                       false, a, false, b0, (short)0, acc0, false, false);
            acc1 = __builtin_amdgcn_wmma_f32_16x16x4_f32(
                       false, a, false, b1, (short)0, acc1, false, false);
        }
        __syncthreads();   // buffer `cur` free before it is DMA'd again
    }
#else
    for (int ch = 0; ch < NCHUNKS; ++ch) {
        const int n0 = n_start + ch * KCHUNK;
        __syncthreads();
        #pragma unroll
        for (int i = 0; i < 16; ++i) {
            int idx = i * 256 + tid;
            int nl  = idx & 63;
            int b   = idx >> 6;
            Xs[0][b][nl] = x[(size_t)(b * N_LEN + n0 + nl) * (C_CH * 2) + c * 2 + p];
        }
        if (tid < 128) {
            int widx = n0 - 63 + tid;
            Ws[0][tid] = (widx >= 0 && widx < WIN_K) ? w_in[c * WIN_K + widx] : 0.0f;
        }
        __syncthreads();
        #pragma unroll
        for (int kk = 0; kk < KCHUNK; kk += 4) {
            const int krow = kk + 2 * hi;
            v2f a = *(const v2f*)&Xs[0][m_base + lm][krow];
            int j0 = krow - (t_base + lm) + 63;
            v2f b0; b0.x = Ws[0][j0];      b0.y = Ws[0][j0 + 1];
            v2f b1; b1.x = Ws[0][j0 - 16]; b1.y = Ws[0][j0 - 15];
            acc0 = __builtin_amdgcn_wmma_f32_16x16x4_f32(
                       false, a, false, b0, (short)0, acc0, false, false);
            acc1 = __builtin_amdgcn_wmma_f32_16x16x4_f32(
                       false, a, false, b1, (short)0, acc1, false, false);
        }
    }
#endif

    // Write partials: part[p][c][ks][b][t]
    float* dst = part + ((size_t)(p * C_CH + c) * KSPLIT + ks) * (B_SZ * W_OUT);
    #pragma unroll
    for (int r = 0; r < 8; ++r) {
        int brow = m_base + r + 8 * hi;   // C/D layout: M = r + 8*hi, N = lm
        dst[brow * W_OUT + t_base + lm]      = acc0[r];
        dst[brow * W_OUT + t_base + 16 + lm] = acc1[r];
    }
}

// One block per channel, thread = batch index. Reduces k-splits, applies the
// per-channel 2x2, cubic amplitude gate, shared 2x2, output filters, and the
// fused training-mode BatchNorm over the 128 (b,part) values of this channel.
__global__ __launch_bounds__(64)
void tail_kernel(const float* __restrict__ part,
                 const float* __restrict__ A_nl,
                 const float* __restrict__ Wc,
                 const float* __restrict__ w_out_real,
                 const float* __restrict__ w_out_imag,
                 const float* __restrict__ gamma,
                 const float* __restrict__ beta,
                 float* __restrict__ out)
{
    const int c = blockIdx.x;
    const int b = threadIdx.x;   // 0..63
    const float a00 = A_nl[c * 4 + 0], a01 = A_nl[c * 4 + 1];
    const float a10 = A_nl[c * 4 + 2], a11 = A_nl[c * 4 + 3];
    const float wc00 = Wc[0], wc01 = Wc[1], wc10 = Wc[2], wc11 = Wc[3];

    const int ksStride = B_SZ * W_OUT;  // 4096
    const float* pr = part + (size_t)(0 * C_CH + c) * KSPLIT * ksStride + b * W_OUT;
    const float* pi = part + (size_t)(1 * C_CH + c) * KSPLIT * ksStride + b * W_OUT;

    float o_r = 0.f, o_i = 0.f;
    for (int t = 0; t < W_OUT; ++t) {
        float fr = pr[t] + pr[ksStride + t] + pr[2 * ksStride + t] + pr[3 * ksStride + t];
        float fi = pi[t] + pi[ksStride + t] + pi[2 * ksStride + t] + pi[3 * ksStride + t];
        float ur = a00 * fr + a01 * fi;
        float ui = a10 * fr + a11 * fi;
        float amp = ur * ur + ui * ui;
        ur *= amp; ui *= amp;
        float sr = wc00 * ur + wc01 * ui;
        float si = wc10 * ur + wc11 * ui;
        o_r += sr * w_out_real[c * W_OUT + t];
        o_i += si * w_out_imag[c * W_OUT + t];
    }

    // BatchNorm stats over (b, part): 128 values per channel
    float s1 = o_r + o_i;
    float s2 = o_r * o_r + o_i * o_i;
    #pragma unroll
    for (int m = 16; m >= 1; m >>= 1) {
        s1 += __shfl_xor(s1, m, 32);
        s2 += __shfl_xor(s2, m, 32);
    }
    __shared__ float red[4];
    const int wv = threadIdx.x >> 5;
    if ((threadIdx.x & 31) == 0) { red[wv * 2] = s1; red[wv * 2 + 1] = s2; }
    __syncthreads();
    float S1 = red[0] + red[2];
    float S2 = red[1] + red[3];
    float mean = S1 * (1.0f / 128.0f);
    float var  = S2 * (1.0f / 128.0f) - mean * mean;
    float scl  = rsqrtf(var + 1e-5f) * gamma[c];
    float sh   = beta[c];
    out[(b * C_CH + c) * 2 + 0] = (o_r - mean) * scl + sh;
    out[(b * C_CH + c) * 2 + 1] = (o_i - mean) * scl + sh;
}

extern "C" void kernel_launch(void* const* d_in, const int* in_sizes, int n_in,
                              void* d_out, int out_size, void* d_ws, size_t ws_size,
                              hipStream_t stream) {
    const float* x   = (const float*)d_in[0];
    const float* wir = (const float*)d_in[1];
    const float* wii = (const float*)d_in[2];
    const float* A   = (const float*)d_in[3];
    const float* Wc  = (const float*)d_in[4];
    const float* wor = (const float*)d_in[5];
    const float* woi = (const float*)d_in[6];
    const float* g   = (const float*)d_in[7];
    const float* be  = (const float*)d_in[8];
    (void)in_sizes; (void)n_in; (void)out_size; (void)ws_size;

    float* part = (float*)d_ws;   // [2][128][KSPLIT][64][64] f32 = 16 MB
    float* out  = (float*)d_out;

    dim3 grid1(C_CH, 2, KSPLIT);  // 1024 blocks x 8 waves
    corr_gemm_kernel<<<grid1, 256, 0, stream>>>(x, wir, wii, part);
    tail_kernel<<<C_CH, 64, 0, stream>>>(part, A, Wc, wor, woi, g, be, out);
}